// PositionalNgramMemoryNetwork_1125281431621
// MI455X (gfx1250) — compile-verified
//
#include <hip/hip_runtime.h>

typedef __attribute__((ext_vector_type(2))) float v2f;
typedef __attribute__((ext_vector_type(8))) float v8f;

#define LSEQ   2048
#define DDIM   256
#define SLOTS  64
#define JCOLS  192              // SLOTS * NGRAM
#define ROWS   8192             // B * L
#define TROWS  64               // rows per workgroup
#define R_TILES 5               // 4 row tiles + 1 halo tile (rows t0-16 .. t0+63)
#define PSTR   196              // padded LDS row stride (floats), 196 % 64 == 4
#define SSTR   196

// LDS layout (floats): P[80][PSTR] | S[64][SSTR] | pos_bias[192]
#define LDS_P_OFF  0
#define LDS_S_OFF  (80 * PSTR)
#define LDS_PB_OFF (LDS_S_OFF + 64 * SSTR)
#define LDS_FLOATS (LDS_PB_OFF + JCOLS)

__global__ __launch_bounds__(256)
void png_mem_net_kernel(const float* __restrict__ x,
                        const float* __restrict__ mem,
                        const float* __restrict__ pb,
                        float* __restrict__ out) {
    extern __shared__ float lds[];
    const int tid   = threadIdx.x;
    const int lane  = tid & 31;
    const int wave  = tid >> 5;
    const int lmod  = lane & 15;      // N / M index within fragment
    const int lhalf = lane >> 4;      // K-pair selector for A/B fragments
    const int t0    = blockIdx.x * TROWS;

    // stage pos_bias into LDS (covered by the phase-1 barrier)
    if (tid < JCOLS) lds[LDS_PB_OFF + tid] = pb[tid];

    // ---------------- Phase 1: P[t,j] = dot(x[t], mem[j]) ----------------
    // rows t in [t0-16, t0+64), cols j in [0,192). 5x12 = 60 WMMA C-tiles.
    for (int tile = wave; tile < R_TILES * 12; tile += 8) {
        const int rt = tile / 12;
        const int ct = tile - rt * 12;
        const int j0 = ct * 16;
        int r = t0 - 16 + rt * 16 + lmod;
        if (r < 0) r = 0;                       // clamp; masked later by validity
        const float* aptr = x   + (size_t)r * DDIM + (lhalf << 1);
        const float* bptr = mem + (size_t)(j0 + lmod) * DDIM + (lhalf << 1);
        v8f acc = {};
        #pragma unroll 8
        for (int k0 = 0; k0 < DDIM; k0 += 4) {
            v2f a = *(const v2f*)(aptr + k0);   // A[lmod, k0 + 2*lhalf + {0,1}]
            v2f b = *(const v2f*)(bptr + k0);   // B[k0 + 2*lhalf + {0,1}, lmod]
            acc = __builtin_amdgcn_wmma_f32_16x16x4_f32(
                false, a, false, b, (short)0, acc, false, false);
        }
        const int rbase = rt * 16 + (lhalf << 3);
        #pragma unroll
        for (int v = 0; v < 8; ++v)
            lds[LDS_P_OFF + (rbase + v) * PSTR + j0 + lmod] = acc[v];
    }
    __syncthreads();

    // ---------------- Phase 2: argmax over n -> one-hot S ----------------
    for (int task = tid; task < TROWS * SLOTS; task += 256) {
        const int i = task >> 6;              // row within block tile
        const int m = task & (SLOTS - 1);     // slot
        const int l = (t0 + i) & (LSEQ - 1);  // position within batch
        // n = 0 reads row t-2 (LDS row i+14), n = 2 reads row t (LDS row i+16)
        float s0 = ((l - 2 >= 0) ? lds[LDS_P_OFF + (i + 14) * PSTR + m * 3 + 0] : 0.f)
                   + lds[LDS_PB_OFF + m * 3 + 0];
        float s1 = ((l - 1 >= 0) ? lds[LDS_P_OFF + (i + 15) * PSTR + m * 3 + 1] : 0.f)
                   + lds[LDS_PB_OFF + m * 3 + 1];
        float s2 = lds[LDS_P_OFF + (i + 16) * PSTR + m * 3 + 2]
                   + lds[LDS_PB_OFF + m * 3 + 2];
        int best = 0; float bs = s0;          // strict '>' => first-max, JAX tie rule
        if (s1 > bs) { bs = s1; best = 1; }
        if (s2 > bs) { best = 2; }
        lds[LDS_S_OFF + i * SSTR + m * 3 + 0] = (best == 0) ? 1.f : 0.f;
        lds[LDS_S_OFF + i * SSTR + m * 3 + 1] = (best == 1) ? 1.f : 0.f;
        lds[LDS_S_OFF + i * SSTR + m * 3 + 2] = (best == 2) ? 1.f : 0.f;
    }
    __syncthreads();

    // ---------------- Phase 3: out = S[64x192] . Mem[192x256] ----------------
    // 4x16 = 64 WMMA C-tiles over (rows, d-columns).
    for (int tile = wave; tile < 4 * 16; tile += 8) {
        const int rt = tile >> 4;
        const int dt = tile & 15;
        const int i0 = rt * 16;
        const int d0 = dt * 16;
        const float* sptr = lds + LDS_S_OFF + (i0 + lmod) * SSTR + (lhalf << 1);
        v8f acc = {};
        #pragma unroll 4
        for (int k0 = 0; k0 < JCOLS; k0 += 4) {
            v2f a = *(const v2f*)(sptr + k0);           // ds_load_b64 of S fragment
            const int kb = k0 + (lhalf << 1);
            v2f b;
            b.x = mem[(size_t)(kb + 0) * DDIM + d0 + lmod];
            b.y = mem[(size_t)(kb + 1) * DDIM + d0 + lmod];
            acc = __builtin_amdgcn_wmma_f32_16x16x4_f32(
                false, a, false, b, (short)0, acc, false, false);
        }
        const int rbase = t0 + i0 + (lhalf << 3);
        #pragma unroll
        for (int v = 0; v < 8; ++v)
            out[(size_t)(rbase + v) * DDIM + d0 + lmod] = acc[v];
    }
}

extern "C" void kernel_launch(void* const* d_in, const int* in_sizes, int n_in,
                              void* d_out, int out_size, void* d_ws, size_t ws_size,
                              hipStream_t stream) {
    (void)in_sizes; (void)n_in; (void)out_size; (void)d_ws; (void)ws_size;
    const float* x   = (const float*)d_in[0];   // (4, 2048, 256) f32
    const float* mem = (const float*)d_in[1];   // (64, 3, 256)  f32
    const float* pb  = (const float*)d_in[2];   // (64, 3)       f32
    float* out = (float*)d_out;                 // (4, 2048, 256) f32

    dim3 grid(ROWS / TROWS);                    // 128 workgroups
    dim3 block(256);                            // 8 wave32 waves
    size_t lds_bytes = (size_t)LDS_FLOATS * sizeof(float);  // 113,664 B < 320 KB
    png_mem_net_kernel<<<grid, block, lds_bytes, stream>>>(x, mem, pb, out);
}